// GCNAutoencoder_4827543241244
// MI455X (gfx1250) — compile-verified
//
#include <hip/hip_runtime.h>
#include <hip/hip_bf16.h>

// ---------------- problem constants ----------------
constexpr int NMm = 100000;      // members
constexpr int NPp = 20000;       // providers
constexpr int NN  = NMm + NPp;   // 120000 nodes (= 16*7500, boundary at 16*6250)
constexpr int DMd = 128;
constexpr int DPd = 64;
constexpr int Ee  = 500000;
constexpr int HIDh = 128;
constexpr int LATl = 32;

// ---------------- WMMA helper (fp32 16x16x4) ----------------
typedef __attribute__((ext_vector_type(2))) float v2f;
typedef __attribute__((ext_vector_type(8))) float v8f;

__device__ __forceinline__ v8f wmma_f32_k4(v2f a, v2f b, v8f c) {
  // 8-arg pattern: (neg_a, A, neg_b, B, c_mod, C, reuse_a, reuse_b)
  return __builtin_amdgcn_wmma_f32_16x16x4_f32(false, a, false, b, (short)0, c,
                                               false, false);
}

__device__ __forceinline__ void atomAddF(float* p, float v) {
  unsafeAtomicAdd(p, v);   // -> global_atomic_add_f32 (no return, STOREcnt)
}

// ---------------- degree / norm ----------------
__global__ void __launch_bounds__(256) k_init_deg(float* __restrict__ deg) {
  int i = blockIdx.x * 256 + threadIdx.x;
  if (i < NN) deg[i] = 1.0f;               // self-loop
}

__global__ void __launch_bounds__(256) k_deg_edges(const int* __restrict__ pidx,
                                                   const int* __restrict__ midx,
                                                   float* __restrict__ deg) {
  int e = blockIdx.x * 256 + threadIdx.x;
  if (e >= Ee) return;
  int d1 = midx[e] + NMm;                  // may be OOB -> scatter dropped (JAX)
  if (d1 < NN) atomAddF(&deg[d1], 1.0f);
  atomAddF(&deg[pidx[e]], 1.0f);           // always in range
}

__global__ void __launch_bounds__(256) k_dinv(float* __restrict__ deg) {
  int i = blockIdx.x * 256 + threadIdx.x;
  if (i < NN) deg[i] = rsqrtf(deg[i]);     // deg^(-1/2), in place
}

// ---------------- GEMM1: xw = X @ W1  (NNx128 * 128x128) ----------------
__global__ void __launch_bounds__(256) k_gemm1(const float* __restrict__ xm,
                                               const float* __restrict__ xp,
                                               const float* __restrict__ W1,
                                               float* __restrict__ xw) {
  __shared__ float Xs[16][132];            // stride 132 % 64 banks = 4
  const int rowBase = blockIdx.x * 16;
  const int t = threadIdx.x;
  for (int i = t; i < 16 * 128; i += 256) {
    int r = i >> 7, c = i & 127;
    int row = rowBase + r;
    float v;
    if (row < NMm) v = xm[(size_t)row * DMd + c];
    else           v = (c < DPd) ? xp[(size_t)(row - NMm) * DPd + c] : 0.0f;
    Xs[r][c] = v;
  }
  __syncthreads();
  const int wave = t >> 5, lane = t & 31;
  const int half = lane >> 4, ml = lane & 15;
  const int colBase = wave * 16;           // 8 waves -> 8 col tiles of 16
  v8f acc = {};
#pragma unroll
  for (int k = 0; k < 128; k += 4) {
    v2f a, b;
    a[0] = Xs[ml][k + 2 * half];
    a[1] = Xs[ml][k + 2 * half + 1];
    b[0] = W1[(size_t)(k + 2 * half)     * HIDh + colBase + ml];
    b[1] = W1[(size_t)(k + 2 * half + 1) * HIDh + colBase + ml];
    acc = wmma_f32_k4(a, b, acc);
  }
#pragma unroll
  for (int i = 0; i < 8; i++) {
    int row = rowBase + half * 8 + i;      // C/D layout: vgpr i = M=i / M=i+8
    xw[(size_t)row * HIDh + colBase + ml] = acc[i];
  }
}

// ---------------- GEMM2: hw = relu(out1 + b1) @ W2  (NNx128 * 128x32) -------
__global__ void __launch_bounds__(256) k_gemm2(const float* __restrict__ out1,
                                               const float* __restrict__ b1,
                                               const float* __restrict__ W2,
                                               float* __restrict__ hw) {
  __shared__ float Hs[64][132];
  const int rowBase = blockIdx.x * 64;
  const int t = threadIdx.x;
  for (int i = t; i < 64 * 128; i += 256) {
    int r = i >> 7, c = i & 127;
    float v = out1[(size_t)(rowBase + r) * HIDh + c] + b1[c];
    Hs[r][c] = fmaxf(v, 0.0f);             // fused bias + ReLU
  }
  __syncthreads();
  const int wave = t >> 5, lane = t & 31;
  const int strip = wave >> 1;             // 4 row strips of 16
  const int colBase = (wave & 1) * 16;     // 2 col tiles
  const int half = lane >> 4, ml = lane & 15;
  v8f acc = {};
#pragma unroll
  for (int k = 0; k < 128; k += 4) {
    v2f a, b;
    a[0] = Hs[strip * 16 + ml][k + 2 * half];
    a[1] = Hs[strip * 16 + ml][k + 2 * half + 1];
    b[0] = W2[(size_t)(k + 2 * half)     * LATl + colBase + ml];
    b[1] = W2[(size_t)(k + 2 * half + 1) * LATl + colBase + ml];
    acc = wmma_f32_k4(a, b, acc);
  }
#pragma unroll
  for (int i = 0; i < 8; i++) {
    int row = rowBase + strip * 16 + half * 8 + i;
    hw[(size_t)row * LATl + colBase + ml] = acc[i];
  }
}

// ---------------- Decoder: out = (out2 + b2) @ Wdec + bdec ------------------
__global__ void __launch_bounds__(256) k_gemmdec(const float* __restrict__ out2,
                                                 const float* __restrict__ b2,
                                                 const float* __restrict__ Wdec,
                                                 const float* __restrict__ bdec,
                                                 float* __restrict__ out) {
  __shared__ float Zs[16][36];
  const int rowBase = blockIdx.x * 16;
  const int t = threadIdx.x;
  for (int i = t; i < 16 * 32; i += 256) {
    int r = i >> 5, c = i & 31;
    Zs[r][c] = out2[(size_t)(rowBase + r) * LATl + c] + b2[c];
  }
  __syncthreads();
  const int wave = t >> 5, lane = t & 31;
  const int colBase = wave * 16;
  const int half = lane >> 4, ml = lane & 15;
  v8f acc = {};
#pragma unroll
  for (int k = 0; k < 32; k += 4) {
    v2f a, b;
    a[0] = Zs[ml][k + 2 * half];
    a[1] = Zs[ml][k + 2 * half + 1];
    b[0] = Wdec[(size_t)(k + 2 * half)     * DMd + colBase + ml];
    b[1] = Wdec[(size_t)(k + 2 * half + 1) * DMd + colBase + ml];
    acc = wmma_f32_k4(a, b, acc);
  }
  const int col = colBase + ml;
  const float bias = bdec[col];
#pragma unroll
  for (int i = 0; i < 8; i++) {
    int row = rowBase + half * 8 + i;
    float v = acc[i] + bias;
    if (row < NMm) out[(size_t)row * DMd + col] = v;                 // member block
    else if (col < DPd)
      out[(size_t)NMm * DMd + (size_t)(row - NMm) * DPd + col] = v;  // provider block
  }
}

// ---------------- self-loop term: acc = dinv^2 * src ------------------------
template <int F>
__global__ void __launch_bounds__(256) k_selfloop(const float* __restrict__ dinv,
                                                  const float* __restrict__ src,
                                                  float* __restrict__ dst) {
  int i = blockIdx.x * 256 + threadIdx.x;
  if (i < NN * F) {
    int node = i / F;
    float d = dinv[node];
    dst[i] = d * d * src[i];
  }
}

// ---------------- edge scatter: acc[d] += dinv[s]*dinv[d]*src[s] ------------
// One wave per edge. F=128: lane owns float4; F=32: lane owns one float.
template <int F>
__global__ void __launch_bounds__(256) k_edge_agg(const int* __restrict__ pidx,
                                                  const int* __restrict__ midx,
                                                  const float* __restrict__ dinv,
                                                  const float* __restrict__ src,
                                                  float* __restrict__ acc) {
  const int lane = threadIdx.x & 31;
  const int wave = (blockIdx.x * 256 + threadIdx.x) >> 5;
  const int nwaves = (gridDim.x * 256) >> 5;
  for (int e = wave; e < Ee; e += nwaves) {
    const int p  = pidx[e];                // in [0, NP) -> always valid node id
    const int mn = midx[e] + NMm;          // may exceed NN (JAX OOB semantics)
    if constexpr (F == 128) {
      const int f = lane * 4;
      // dir A: s = p, d = mn  (drop scatter if d OOB)
      if (mn < NN) {
        float nrm = dinv[p] * dinv[mn];
        const float4 v = *(const float4*)(src + (size_t)p * F + f);
        float* b = acc + (size_t)mn * F + f;
        atomAddF(b + 0, nrm * v.x); atomAddF(b + 1, nrm * v.y);
        atomAddF(b + 2, nrm * v.z); atomAddF(b + 3, nrm * v.w);
      }
      // dir B: s = clamp(mn), d = p  (gather clamps)
      {
        const int s = (mn < NN) ? mn : NN - 1;
        float nrm = dinv[s] * dinv[p];
        const float4 v = *(const float4*)(src + (size_t)s * F + f);
        float* b = acc + (size_t)p * F + f;
        atomAddF(b + 0, nrm * v.x); atomAddF(b + 1, nrm * v.y);
        atomAddF(b + 2, nrm * v.z); atomAddF(b + 3, nrm * v.w);
      }
    } else {  // F == 32
      const int f = lane;
      if (mn < NN) {
        float nrm = dinv[p] * dinv[mn];
        atomAddF(acc + (size_t)mn * F + f, nrm * src[(size_t)p * F + f]);
      }
      {
        const int s = (mn < NN) ? mn : NN - 1;
        float nrm = dinv[s] * dinv[p];
        atomAddF(acc + (size_t)p * F + f, nrm * src[(size_t)s * F + f]);
      }
    }
  }
}

// ---------------- edge_logits = 0 ----------------
__global__ void __launch_bounds__(256) k_zero_logits(float* __restrict__ lg) {
  int i = blockIdx.x * 256 + threadIdx.x;
  if (i < Ee) lg[i] = 0.0f;
}

// ---------------- launch ----------------
extern "C" void kernel_launch(void* const* d_in, const int* in_sizes, int n_in,
                              void* d_out, int out_size, void* d_ws, size_t ws_size,
                              hipStream_t stream) {
  const float* xm   = (const float*)d_in[0];
  const float* xp   = (const float*)d_in[1];
  const int*   pidx = (const int*)d_in[2];
  const int*   midx = (const int*)d_in[3];
  const float* W1   = (const float*)d_in[4];
  const float* b1   = (const float*)d_in[5];
  const float* W2   = (const float*)d_in[6];
  const float* b2   = (const float*)d_in[7];
  const float* Wdec = (const float*)d_in[8];
  const float* bdec = (const float*)d_in[9];
  float* out = (float*)d_out;

  // workspace layout (floats). xw region is reused for hw/out2 once dead.
  float* ws   = (float*)d_ws;
  float* dinv = ws;                            // NN
  float* xw   = ws + NN;                       // NN*128  (conv1 X@W1)
  float* out1 = xw + (size_t)NN * 128;         // NN*128  (conv1 aggregate)
  float* hw   = xw;                            // NN*32   (conv2 H@W2, aliases dead xw)
  float* out2 = xw + (size_t)NN * 32;          // NN*32   (conv2 aggregate)

  // degrees + norm
  k_init_deg<<<(NN + 255) / 256, 256, 0, stream>>>(dinv);
  k_deg_edges<<<(Ee + 255) / 256, 256, 0, stream>>>(pidx, midx, dinv);
  k_dinv<<<(NN + 255) / 256, 256, 0, stream>>>(dinv);

  // conv1
  k_gemm1<<<NN / 16, 256, 0, stream>>>(xm, xp, W1, xw);
  k_selfloop<128><<<(NN * 128 + 255) / 256, 256, 0, stream>>>(dinv, xw, out1);
  k_edge_agg<128><<<4096, 256, 0, stream>>>(pidx, midx, dinv, xw, out1);

  // conv2 (bias+ReLU fused into GEMM2 LDS stage)
  k_gemm2<<<NN / 64, 256, 0, stream>>>(out1, b1, W2, hw);
  k_selfloop<32><<<(NN * 32 + 255) / 256, 256, 0, stream>>>(dinv, hw, out2);
  k_edge_agg<32><<<4096, 256, 0, stream>>>(pidx, midx, dinv, hw, out2);

  // decoder (+b2 fused into LDS stage, +bdec fused into store, split outputs)
  k_gemmdec<<<NN / 16, 256, 0, stream>>>(out2, b2, Wdec, bdec, out);

  // edge_logits = zeros
  k_zero_logits<<<(Ee + 255) / 256, 256, 0, stream>>>(
      out + (size_t)NMm * DMd + (size_t)NPp * DPd);
}